// MultiBoxLoss_88596585382147
// MI455X (gfx1250) — compile-verified
//
#include <hip/hip_runtime.h>
#include <hip/hip_bf16.h>

#define BATCH   128
#define NPRI    8732
#define NCLS    21
#define NOBJ    24
#define NSORT   16384   // next pow2 >= NPRI, for bitonic sort
#define THR     0.5f
#define NEGPOS  3

typedef __attribute__((ext_vector_type(2))) float v2f;
typedef __attribute__((ext_vector_type(8))) float v8f;

// ---------------------------------------------------------------------------
// Kernel 1: per-image matching + localization L1 partial + n_pos
// ---------------------------------------------------------------------------
__global__ __launch_bounds__(256)
void match_kernel(const float* __restrict__ plocs,
                  const float* __restrict__ boxes,
                  const int*   __restrict__ labels,
                  const float* __restrict__ priors,
                  int*   __restrict__ tc,        // [B*NPRI] true class per prior
                  int*   __restrict__ npos_g,    // [B]
                  float* __restrict__ locpart)   // [B]
{
    __shared__ float soverlap[NPRI];
    __shared__ int   sobj[NPRI];
    __shared__ unsigned long long skey[NOBJ];
    __shared__ float bx1[NOBJ], by1[NOBJ], bx2[NOBJ], by2[NOBJ], barea[NOBJ];
    __shared__ int   slab[NOBJ];
    __shared__ float redf[256];
    __shared__ int   redi[256];

    const int b   = blockIdx.x;
    const int tid = threadIdx.x;

    if (tid < NOBJ) {
        float x1 = boxes[((size_t)b * NOBJ + tid) * 4 + 0];
        float y1 = boxes[((size_t)b * NOBJ + tid) * 4 + 1];
        float x2 = boxes[((size_t)b * NOBJ + tid) * 4 + 2];
        float y2 = boxes[((size_t)b * NOBJ + tid) * 4 + 3];
        bx1[tid] = x1; by1[tid] = y1; bx2[tid] = x2; by2[tid] = y2;
        barea[tid] = (x2 - x1) * (y2 - y1);
        slab[tid]  = labels[b * NOBJ + tid];
        skey[tid]  = 0ull;
    }
    __syncthreads();

    // per-prior best object (first-max) + per-object best prior (atomic key)
    for (int p = tid; p < NPRI; p += 256) {
        float cx = priors[p * 4 + 0], cy = priors[p * 4 + 1];
        float pw = priors[p * 4 + 2], ph = priors[p * 4 + 3];
        float px1 = cx - pw * 0.5f, py1 = cy - ph * 0.5f;
        float px2 = cx + pw * 0.5f, py2 = cy + ph * 0.5f;
        float pa  = (px2 - px1) * (py2 - py1);
        float bestv = -1.0f; int bestj = 0;
        #pragma unroll 4
        for (int j = 0; j < NOBJ; ++j) {
            float lx = fmaxf(bx1[j], px1), ly = fmaxf(by1[j], py1);
            float hx = fminf(bx2[j], px2), hy = fminf(by2[j], py2);
            float iw = fmaxf(hx - lx, 0.0f), ih = fmaxf(hy - ly, 0.0f);
            float inter = iw * ih;
            float v = inter / (barea[j] + pa - inter);
            if (v > bestv) { bestv = v; bestj = j; }
            unsigned long long key =
                ((unsigned long long)__float_as_uint(v) << 32) |
                (unsigned long long)(0xFFFFFFFFu - (unsigned)p);
            atomicMax(&skey[j], key);
        }
        soverlap[p] = bestv;
        sobj[p]     = bestj;
    }
    __syncthreads();

    // force each object to own its best prior (in-order, later object wins)
    if (tid == 0) {
        for (int j = 0; j < NOBJ; ++j) {
            unsigned int pstar = 0xFFFFFFFFu - (unsigned int)(skey[j] & 0xFFFFFFFFull);
            sobj[pstar]     = j;
            soverlap[pstar] = 1.0f;
        }
    }
    __syncthreads();

    float locsum = 0.0f; int np = 0;
    for (int p = tid; p < NPRI; p += 256) {
        int   j  = sobj[p];
        float ov = soverlap[p];
        int lab  = (ov < THR) ? 0 : slab[j];
        tc[(size_t)b * NPRI + p] = lab;
        if (lab != 0) {
            np++;
            float cx = priors[p * 4 + 0], cy = priors[p * 4 + 1];
            float pw = priors[p * 4 + 2], ph = priors[p * 4 + 3];
            float bcx = (bx1[j] + bx2[j]) * 0.5f;
            float bcy = (by1[j] + by2[j]) * 0.5f;
            float bw  = bx2[j] - bx1[j];
            float bh  = by2[j] - by1[j];
            float g0 = (bcx - cx) / (pw / 10.0f);
            float g1 = (bcy - cy) / (ph / 10.0f);
            float g2 = logf(bw / pw) * 5.0f;
            float g3 = logf(bh / ph) * 5.0f;
            const float* pl = plocs + ((size_t)b * NPRI + p) * 4;
            locsum += fabsf(pl[0] - g0) + fabsf(pl[1] - g1)
                    + fabsf(pl[2] - g2) + fabsf(pl[3] - g3);
        }
    }
    redf[tid] = locsum; redi[tid] = np;
    __syncthreads();
    for (int s = 128; s > 0; s >>= 1) {
        if (tid < s) { redf[tid] += redf[tid + s]; redi[tid] += redi[tid + s]; }
        __syncthreads();
    }
    if (tid == 0) { npos_g[b] = redi[0]; locpart[b] = redf[0]; }
}

// ---------------------------------------------------------------------------
// Kernel 2: per-prior cross-entropy; conf_neg array + positive-conf partials
// grid = (NSORT/256, BATCH)
// ---------------------------------------------------------------------------
__global__ __launch_bounds__(256)
void conf_kernel(const float* __restrict__ scores,
                 const int*   __restrict__ tc,
                 float* __restrict__ conf_neg,    // [B*NSORT], zero padded
                 float* __restrict__ cpos_part)   // [B * gridDim.x]
{
    __shared__ float red[256];
    const int b = blockIdx.y;
    const int p = blockIdx.x * 256 + threadIdx.x;   // < NSORT by construction

    float possum = 0.0f;
    float negval = 0.0f;
    if (p < NPRI) {
        const float* sp = scores + ((size_t)b * NPRI + p) * NCLS;
        float r[NCLS];
        float m = -INFINITY;
        #pragma unroll
        for (int c = 0; c < NCLS; ++c) { r[c] = sp[c]; m = fmaxf(m, r[c]); }
        float se = 0.0f;
        #pragma unroll
        for (int c = 0; c < NCLS; ++c) se += expf(r[c] - m);
        float lse = m + logf(se);
        int lab   = tc[(size_t)b * NPRI + p];
        float conf = lse - r[lab];
        if (lab != 0) possum = conf; else negval = conf;
    }
    conf_neg[(size_t)b * NSORT + p] = negval;

    red[threadIdx.x] = possum;
    __syncthreads();
    for (int s = 128; s > 0; s >>= 1) {
        if (threadIdx.x < s) red[threadIdx.x] += red[threadIdx.x + s];
        __syncthreads();
    }
    if (threadIdx.x == 0) cpos_part[b * gridDim.x + blockIdx.x] = red[0];
}

// ---------------------------------------------------------------------------
// Kernel 3: per-image bitonic sort (descending) + top-(3*n_pos) sum
// ---------------------------------------------------------------------------
__global__ __launch_bounds__(1024)
void hardneg_kernel(const float* __restrict__ conf_neg,
                    const int*   __restrict__ npos_g,
                    float* __restrict__ chard)    // [B]
{
    __shared__ float buf[NSORT];
    __shared__ float red[1024];
    const int b = blockIdx.x, tid = threadIdx.x;

    for (int i = tid; i < NSORT; i += 1024)
        buf[i] = conf_neg[(size_t)b * NSORT + i];
    __syncthreads();

    for (int k = 2; k <= NSORT; k <<= 1) {
        for (int j = k >> 1; j > 0; j >>= 1) {
            for (int i = tid; i < NSORT; i += 1024) {
                int ixj = i ^ j;
                if (ixj > i) {
                    float a = buf[i], c = buf[ixj];
                    bool down = ((i & k) == 0);           // descending overall
                    bool sw = down ? (a < c) : (a > c);
                    if (sw) { buf[i] = c; buf[ixj] = a; }
                }
            }
            __syncthreads();
        }
    }

    int nh = NEGPOS * npos_g[b];
    int kk = nh < NSORT ? nh : NSORT;
    float sum = 0.0f;
    for (int i = tid; i < kk; i += 1024) sum += buf[i];
    red[tid] = sum;
    __syncthreads();
    for (int s = 512; s > 0; s >>= 1) {
        if (tid < s) red[tid] += red[tid + s];
        __syncthreads();
    }
    if (tid == 0) chard[b] = red[0];
}

// ---------------------------------------------------------------------------
// Kernel 4: exact fp32 sums via V_WMMA_F32_16X16X4_F32 (A = ones)
// D[i][j] = sum_k B[k][j]  -> total = sum over D row 0 (lanes 0..15, VGPR0).
// Total is invariant to the exact B VGPR layout: all 64 slots are filled and
// summation is permutation-invariant.
// ---------------------------------------------------------------------------
__device__ inline float wmma_sum(const float* __restrict__ data, int chunks, int lane)
{
    v2f a; a.x = 1.0f; a.y = 1.0f;
    v8f c = {};
    for (int t = 0; t < chunks; ++t) {
        v2f bv;
        bv.x = data[t * 64 + lane];
        bv.y = data[t * 64 + lane + 32];
        c = __builtin_amdgcn_wmma_f32_16x16x4_f32(false, a, false, bv,
                                                  (short)0, c, false, false);
    }
    float v = c[0];                 // D(M = 0 or 8, N = lane%16) = colsum_N
    v += __shfl_xor(v, 1);
    v += __shfl_xor(v, 2);
    v += __shfl_xor(v, 4);
    v += __shfl_xor(v, 8);          // lane 0: sum of colsums 0..15 = total
    return v;
}

__global__ __launch_bounds__(32)
void final_kernel(const float* __restrict__ locpart,
                  const int*   __restrict__ npos_g,
                  const float* __restrict__ chard,
                  const float* __restrict__ cpos_part,
                  int ncpos,
                  float* __restrict__ out)
{
    __shared__ float tmp[BATCH];
    const int lane = threadIdx.x;

    float loc_total   = wmma_sum(locpart,   BATCH / 64, lane);
    float chard_total = wmma_sum(chard,     BATCH / 64, lane);
    float cpos_total  = wmma_sum(cpos_part, ncpos / 64, lane);

    for (int i = lane; i < BATCH; i += 32) tmp[i] = (float)npos_g[i];
    __syncthreads();
    float tp = wmma_sum(tmp, BATCH / 64, lane);

    if (lane == 0) {
        float loc_loss  = loc_total / (tp * 4.0f);
        float conf_loss = (cpos_total + chard_total) / tp;
        out[0] = conf_loss + loc_loss;
    }
}

// ---------------------------------------------------------------------------
// Launch
// ---------------------------------------------------------------------------
extern "C" void kernel_launch(void* const* d_in, const int* in_sizes, int n_in,
                              void* d_out, int out_size, void* d_ws, size_t ws_size,
                              hipStream_t stream) {
    const float* plocs  = (const float*)d_in[0];   // (B, N, 4)
    const float* scores = (const float*)d_in[1];   // (B, N, 21)
    const float* boxes  = (const float*)d_in[2];   // (B, 24, 4)
    const int*   labels = (const int*)  d_in[3];   // (B, 24)
    const float* priors = (const float*)d_in[4];   // (N, 4)
    float* out = (float*)d_out;

    char* ws = (char*)d_ws;
    size_t off = 0;
    int*   tc        = (int*)  (ws + off); off += (size_t)BATCH * NPRI * 4;   // 4.47 MB
    float* conf_neg  = (float*)(ws + off); off += (size_t)BATCH * NSORT * 4;  // 8.39 MB
    int*   npos_g    = (int*)  (ws + off); off += 512;
    float* locpart   = (float*)(ws + off); off += 512;
    float* chard     = (float*)(ws + off); off += 512;
    const int NBLK = NSORT / 256;                                  // 64 blocks/image
    float* cpos_part = (float*)(ws + off); off += (size_t)BATCH * NBLK * 4;

    match_kernel<<<BATCH, 256, 0, stream>>>(plocs, boxes, labels, priors,
                                            tc, npos_g, locpart);
    conf_kernel<<<dim3(NBLK, BATCH), 256, 0, stream>>>(scores, tc,
                                                       conf_neg, cpos_part);
    hardneg_kernel<<<BATCH, 1024, 0, stream>>>(conf_neg, npos_g, chard);
    final_kernel<<<1, 32, 0, stream>>>(locpart, npos_g, chard, cpos_part,
                                       BATCH * NBLK, out);
}